// ViTBlockQuantum_22582938042552
// MI455X (gfx1250) — compile-verified
//
#include <hip/hip_runtime.h>
#include <math.h>

// Problem constants (from reference setup_inputs).
#define B_DIM 64
#define S_DIM 256
#define E_DIM 8
#define H_DIM 2
#define DK    4          // E / H
#define EPS   1e-5f
// 0.5f (1/sqrt(dk)) folded with log2(e) for the v_exp_f32 path.
#define SCALE_LOG2E 0.72134752044448170f

typedef __attribute__((ext_vector_type(2))) float v2f;
typedef __attribute__((ext_vector_type(8))) float v8f;

// ---------------------------------------------------------------------------
// DPP16 butterfly add across each 16-lane half (the row group of the WMMA
// C/D layout). xor1 = quad_perm[1,0,3,2], xor2 = quad_perm[2,3,0,1],
// then row_half_mirror (== xor4 once quads are uniform) and row_mirror
// (== xor8 once 8-groups are uniform). Pure VALU, no LDS.
// ---------------------------------------------------------------------------
template <int CTRL>
__device__ __forceinline__ float dpp_f(float x) {
  return __int_as_float(__builtin_amdgcn_update_dpp(
      0, __float_as_int(x), CTRL, 0xF, 0xF, true));
}
__device__ __forceinline__ float rowsum16(float v) {
  v += dpp_f<0xB1>(v);    // quad_perm [1,0,3,2]
  v += dpp_f<0x4E>(v);    // quad_perm [2,3,0,1]
  v += dpp_f<0x141>(v);   // row_half_mirror
  v += dpp_f<0x140>(v);   // row_mirror
  return v;
}

// ---------------------------------------------------------------------------
// VQC collapse: one RX layer + CNOT ring ==> prefix products of cosines:
//   z_w = prod_{j=0..w} cos(x_j+W_j)  (w>=1),  z_0 = prod_{j=1..7} cos(.)
// ---------------------------------------------------------------------------
__device__ __forceinline__ void vqc8(const float (&in)[8], const float (&W)[8],
                                     float (&z)[8]) {
  float c[8];
#pragma unroll
  for (int j = 0; j < 8; ++j) c[j] = __cosf(in[j] + W[j]);
  float p = c[0];
#pragma unroll
  for (int j = 1; j < 8; ++j) { p *= c[j]; z[j] = p; }
  float q = c[1];
#pragma unroll
  for (int j = 2; j < 8; ++j) q *= c[j];
  z[0] = q;
}

__device__ __forceinline__ void ln8(const float (&t)[8],
                                    const float* __restrict__ g,
                                    const float* __restrict__ b,
                                    float (&o)[8]) {
  float m = 0.f;
#pragma unroll
  for (int j = 0; j < 8; ++j) m += t[j];
  m *= 0.125f;
  float v = 0.f;
#pragma unroll
  for (int j = 0; j < 8; ++j) { float d = t[j] - m; v += d * d; }
  v *= 0.125f;
  float inv = 1.0f / sqrtf(v + EPS);
#pragma unroll
  for (int j = 0; j < 8; ++j) o[j] = (t[j] - m) * inv * g[j] + b[j];
}

// ---------------------------------------------------------------------------
// Kernel A: one block per (batch, head). Stage 1: VQC -> Q/K/V head slices in
// LDS. Stage 2: attention with QK^T on V_WMMA_F32_16X16X4_F32.
// Scores are bounded (|s| <= 2), so softmax needs no max subtraction; the
// denominator stays lane-distributed and is reduced once at the end with DPP.
// ---------------------------------------------------------------------------
__global__ __launch_bounds__(256) void attn_kernel(
    const float* __restrict__ x, const float* __restrict__ Wq,
    const float* __restrict__ Wk, const float* __restrict__ Wv,
    float* __restrict__ ctx) {
  __shared__ __align__(16) float Qs[S_DIM * DK];
  __shared__ __align__(16) float Ks[S_DIM * DK];
  __shared__ __align__(16) float Vs[S_DIM * DK];

  const int b = blockIdx.x >> 1;
  const int h = blockIdx.x & 1;
  const int tid = threadIdx.x;

  float wq[8], wk[8], wv[8];
#pragma unroll
  for (int j = 0; j < 8; ++j) { wq[j] = Wq[j]; wk[j] = Wk[j]; wv[j] = Wv[j]; }

  // ---- Stage 1: per-token VQC (prefix cosines), keep this head's dk slice.
  {
    const int s = tid;  // 256 threads == 256 tokens
    const float* xp = x + (size_t)(b * S_DIM + s) * E_DIM;
    float xr[8];
#pragma unroll
    for (int j = 0; j < 8; ++j) xr[j] = xp[j];
    float zq[8], zk[8], zv[8];
    vqc8(xr, wq, zq);
    vqc8(xr, wk, zk);
    vqc8(xr, wv, zv);
#pragma unroll
    for (int d = 0; d < DK; ++d) {
      Qs[s * DK + d] = zq[h * DK + d];
      Ks[s * DK + d] = zk[h * DK + d];
      Vs[s * DK + d] = zv[h * DK + d];
    }
  }
  __syncthreads();

  // ---- Stage 2: attention, 8 waves x 2 query blocks each.
  const int lane  = tid & 31;
  const int wv_id = tid >> 5;
  const int half  = lane >> 4;   // 0: lanes 0-15, 1: lanes 16-31
  const int ln    = lane & 15;
  const int kp    = half * 2;    // K-pair this half supplies (A/B layout)

  for (int i = 0; i < 2; ++i) {
    const int qb = wv_id * 2 + i;
    const int qbase = qb * 16;

    // A-matrix: lane ln holds Q[qbase+ln][kp], Q[qbase+ln][kp+1]
    v2f a;
    a.x = Qs[(qbase + ln) * DK + kp];
    a.y = Qs[(qbase + ln) * DK + kp + 1];

    float lrow[8], acc[8][DK];
#pragma unroll
    for (int r = 0; r < 8; ++r) {
      lrow[r] = 0.f;
#pragma unroll
      for (int d = 0; d < DK; ++d) acc[r][d] = 0.f;
    }

#pragma unroll
    for (int t = 0; t < 16; ++t) {
      const int kbase = t * 16;
      // B-matrix: lane ln is column N (key index), same K striping as A.
      v2f bm;
      bm.x = Ks[(kbase + ln) * DK + kp];
      bm.y = Ks[(kbase + ln) * DK + kp + 1];

      v8f cz = {0.f, 0.f, 0.f, 0.f, 0.f, 0.f, 0.f, 0.f};
      // D[r] = scores[m = r + 8*half][n = ln]
      v8f d = __builtin_amdgcn_wmma_f32_16x16x4_f32(
          false, a, false, bm, (short)0, cz, false, false);

      float vvals[DK];
#pragma unroll
      for (int dv = 0; dv < DK; ++dv) vvals[dv] = Vs[(kbase + ln) * DK + dv];

#pragma unroll
      for (int r = 0; r < 8; ++r) {
        // exp(score/sqrt(dk)) via single v_exp_f32; |score|<=2 so this is safe.
        float p = __builtin_amdgcn_exp2f(d[r] * SCALE_LOG2E);
        lrow[r] += p;  // lane-distributed denominator (summed over this n)
#pragma unroll
        for (int dv = 0; dv < DK; ++dv) acc[r][dv] += p * vvals[dv];
      }
    }

    // Finalize: DPP butterfly across the 16 n-lanes, normalize, store ctx.
#pragma unroll
    for (int r = 0; r < 8; ++r) {
      float inv_l = 1.0f / rowsum16(lrow[r]);
#pragma unroll
      for (int dv = 0; dv < DK; ++dv) {
        float v = rowsum16(acc[r][dv]) * inv_l;
        if (ln == dv) {
          const int m = qbase + r + 8 * half;
          ctx[(size_t)(b * S_DIM + m) * E_DIM + h * DK + dv] = v;
        }
      }
    }
  }
}

// ---------------------------------------------------------------------------
// Kernel B: per-token epilogue. VQC(ctx,Wc) -> +x -> LN1 -> w1 -> VQC(.,Wf)
// -> w2 -> +LN1 -> LN2 -> out.
// ---------------------------------------------------------------------------
__global__ __launch_bounds__(256) void post_kernel(
    const float* __restrict__ x, const float* __restrict__ ctx,
    const float* __restrict__ Wc, const float* __restrict__ Wf,
    const float* __restrict__ w1, const float* __restrict__ b1,
    const float* __restrict__ w2, const float* __restrict__ b2,
    const float* __restrict__ g1, const float* __restrict__ be1,
    const float* __restrict__ g2, const float* __restrict__ be2,
    float* __restrict__ out) {
  const int gid = blockIdx.x * blockDim.x + threadIdx.x;
  if (gid >= B_DIM * S_DIM) return;

  float xv[8], cv[8], wc[8], wf[8];
#pragma unroll
  for (int j = 0; j < 8; ++j) {
    xv[j] = x[(size_t)gid * E_DIM + j];
    cv[j] = ctx[(size_t)gid * E_DIM + j];
    wc[j] = Wc[j];
    wf[j] = Wf[j];
  }

  float ao[8];
  vqc8(cv, wc, ao);
  float t[8];
#pragma unroll
  for (int j = 0; j < 8; ++j) t[j] = xv[j] + ao[j];
  float y[8];
  ln8(t, g1, be1, y);

  float hv[8];
#pragma unroll
  for (int i = 0; i < 8; ++i) {
    float s = b1[i];
#pragma unroll
    for (int j = 0; j < 8; ++j) s += y[j] * w1[i * 8 + j];  // y @ w1.T
    hv[i] = s;
  }
  float hq[8];
  vqc8(hv, wf, hq);

  float u[8];
#pragma unroll
  for (int i = 0; i < 8; ++i) {
    float s = b2[i];
#pragma unroll
    for (int j = 0; j < 8; ++j) s += hq[j] * w2[i * 8 + j];  // hq @ w2.T
    u[i] = y[i] + s;
  }
  float o[8];
  ln8(u, g2, be2, o);
#pragma unroll
  for (int j = 0; j < 8; ++j) out[(size_t)gid * E_DIM + j] = o[j];
}

extern "C" void kernel_launch(void* const* d_in, const int* in_sizes, int n_in,
                              void* d_out, int out_size, void* d_ws,
                              size_t ws_size, hipStream_t stream) {
  (void)in_sizes; (void)n_in; (void)out_size; (void)ws_size;
  const float* x   = (const float*)d_in[0];
  const float* Wq  = (const float*)d_in[1];
  const float* Wk  = (const float*)d_in[2];
  const float* Wv  = (const float*)d_in[3];
  const float* Wc  = (const float*)d_in[4];
  const float* Wf  = (const float*)d_in[5];
  const float* w1  = (const float*)d_in[6];
  const float* b1  = (const float*)d_in[7];
  const float* w2  = (const float*)d_in[8];
  const float* b2  = (const float*)d_in[9];
  const float* g1  = (const float*)d_in[10];
  const float* be1 = (const float*)d_in[11];
  const float* g2  = (const float*)d_in[12];
  const float* be2 = (const float*)d_in[13];
  float* out = (float*)d_out;
  float* ctx = (float*)d_ws;  // B*S*E floats = 512 KB scratch

  attn_kernel<<<B_DIM * H_DIM, 256, 0, stream>>>(x, Wq, Wk, Wv, ctx);
  post_kernel<<<(B_DIM * S_DIM) / 256, 256, 0, stream>>>(
      x, ctx, Wc, Wf, w1, b1, w2, b2, g1, be1, g2, be2, out);
}